// GRUDecoder_69887707840784
// MI455X (gfx1250) — compile-verified
//
#include <hip/hip_runtime.h>
#include <hip/hip_bf16.h>

// ---------------------------------------------------------------------------
// GNN (3x GraphConv) + mean-pool + 2-layer GRU + sigmoid decoder for gfx1250.
// Dense math on V_WMMA_F32_16X16X4_F32 (exact fp32), all K/F compile-time so
// inner loops are guard-free, fully unrolled, with v2f (b64) fragment loads.
// Edge scatter via global_atomic_add_f32 (agg buffers are L2-resident).
// ---------------------------------------------------------------------------

typedef __attribute__((ext_vector_type(2))) float v2f;
typedef __attribute__((ext_vector_type(8))) float v8f;

#define N_NODES 262144
#define N_EDGES 2097152
#define N_GRAPHS 8192
#define BATCH 512
#define TSTEPS 16
#define HID 128
#define GATE3 384  // 3*HID

__device__ __forceinline__ v8f wmma_f32(v2f a, v2f b, v8f c) {
  // 8 args: (neg_a, A, neg_b, B, c_mod, C, reuse_a, reuse_b)
  return __builtin_amdgcn_wmma_f32_16x16x4_f32(false, a, false, b, (short)0, c,
                                               false, false);
}

__device__ __forceinline__ float sigm(float x) {
  return 1.0f / (1.0f + __expf(-x));
}

// ---------------------------------------------------------------------------
// Layer-1 padding helpers (K=5 -> K=8 so WMMA loops are guard-free)
// ---------------------------------------------------------------------------
__global__ __launch_bounds__(256) void pad_x0(const float* __restrict__ x0,
                                              float* __restrict__ xp, int N) {
  int idx = blockIdx.x * blockDim.x + threadIdx.x;
  if (idx >= N * 8) return;
  int n = idx >> 3, f = idx & 7;
  xp[idx] = (f < 5) ? x0[n * 5 + f] : 0.0f;
}

// pad 5x32 weight -> 8x32 (same row-major [K x F] layout, zero rows 5..7)
__global__ __launch_bounds__(256) void pad_w5(const float* __restrict__ w,
                                              float* __restrict__ wp) {
  int idx = blockIdx.x * blockDim.x + threadIdx.x;
  if (idx >= 8 * 32) return;
  wp[idx] = ((idx >> 5) < 5) ? w[idx] : 0.0f;
}

// ---------------------------------------------------------------------------
// Edge scatter: agg[dst[e]] += ew[e] * x[src[e]]
// Layer-1 form: x is the padded [N x 8] buffer, agg stride 8 (5 live lanes).
// ---------------------------------------------------------------------------
__global__ __launch_bounds__(256) void scatter_f5(
    const float* __restrict__ x, const int* __restrict__ src,
    const int* __restrict__ dst, const float* __restrict__ ew,
    float* __restrict__ agg, int E) {
  int e = blockIdx.x * blockDim.x + threadIdx.x;
  if (e >= E) return;
  float w = ew[e];
  const float* xp = x + (size_t)src[e] * 8;
  float* ap = agg + (size_t)dst[e] * 8;
  float4 v = *(const float4*)xp;
  float v4 = xp[4];
  atomicAdd(ap + 0, w * v.x);
  atomicAdd(ap + 1, w * v.y);
  atomicAdd(ap + 2, w * v.z);
  atomicAdd(ap + 3, w * v.w);
  atomicAdd(ap + 4, w * v4);
}

// F multiple of 4: one thread per (edge, 4-feature chunk); float4 gather.
__global__ __launch_bounds__(256) void scatter_vec4(
    const float* __restrict__ x, const int* __restrict__ src,
    const int* __restrict__ dst, const float* __restrict__ ew,
    float* __restrict__ agg, int E, int F) {
  unsigned idx = blockIdx.x * blockDim.x + threadIdx.x;
  unsigned chunks = (unsigned)F >> 2;
  if (idx >= (unsigned)E * chunks) return;
  unsigned e = idx / chunks;
  unsigned c = idx - e * chunks;
  float w = ew[e];
  float4 v = *((const float4*)(x + (size_t)src[e] * F) + c);
  float* ap = agg + (size_t)dst[e] * F + c * 4;
  atomicAdd(ap + 0, w * v.x);
  atomicAdd(ap + 1, w * v.y);
  atomicAdd(ap + 2, w * v.z);
  atomicAdd(ap + 3, w * v.w);
}

// ---------------------------------------------------------------------------
// GCN dense part: Out = relu(X @ Wroot + AGG @ Wrel + b)
// X,AGG: [N x K]; Wroot,Wrel: [K x F] row-major.
// One wave per 16-row band; A fragments preloaded, reused for all col tiles.
// ---------------------------------------------------------------------------
template <int K, int F>
__global__ __launch_bounds__(256) void gcn_gemm_relu_t(
    const float* __restrict__ X, const float* __restrict__ AGG,
    const float* __restrict__ Wroot, const float* __restrict__ Wrel,
    const float* __restrict__ bias, float* __restrict__ Out, int N) {
  int wave = (int)((blockIdx.x * blockDim.x + threadIdx.x) >> 5);
  int lane = threadIdx.x & 31;
  int row0 = wave << 4;
  if (row0 >= N) return;

  int m = lane & 15;        // A row / B col / C col
  int kbh = lane >> 4;      // which k-pair of the 4-wide k step
  constexpr int KF = K / 4;

  const v2f* xr = (const v2f*)(X + (size_t)(row0 + m) * K);
  const v2f* ar = (const v2f*)(AGG + (size_t)(row0 + m) * K);
  v2f ax[KF], ag[KF];
#pragma unroll
  for (int kk = 0; kk < KF; ++kk) {
    ax[kk] = xr[2 * kk + kbh];
    ag[kk] = ar[2 * kk + kbh];
  }
  int rb = kbh << 3;

  for (int ct = 0; ct < F / 16; ++ct) {
    int col = (ct << 4) + m;
    v8f acc = {};
#pragma unroll
    for (int kk = 0; kk < KF; ++kk) {
      int k = 4 * kk + 2 * kbh;
      v2f b;
      b.x = Wroot[(size_t)k * F + col];
      b.y = Wroot[(size_t)(k + 1) * F + col];
      acc = wmma_f32(ax[kk], b, acc);
    }
#pragma unroll
    for (int kk = 0; kk < KF; ++kk) {
      int k = 4 * kk + 2 * kbh;
      v2f b;
      b.x = Wrel[(size_t)k * F + col];
      b.y = Wrel[(size_t)(k + 1) * F + col];
      acc = wmma_f32(ag[kk], b, acc);
    }
    float bv = bias[col];
#pragma unroll
    for (int v = 0; v < 8; ++v)
      Out[(size_t)(row0 + rb + v) * F + col] = fmaxf(acc[v] + bv, 0.0f);
  }
}

// ---------------------------------------------------------------------------
// GRU input projection: Out[M x 384] = A[M x K] @ Wt[384 x K]^T + bias.
// Wave owns one 16-row band and a group of 6 column tiles (4 groups).
// ---------------------------------------------------------------------------
template <int K>
__global__ __launch_bounds__(256) void gru_xw_gemm(
    const float* __restrict__ A, const float* __restrict__ Wt,
    const float* __restrict__ bias, float* __restrict__ Out, int M) {
  int wave = (int)((blockIdx.x * blockDim.x + threadIdx.x) >> 5);
  int lane = threadIdx.x & 31;
  int rowTile = wave >> 2;
  int cg = wave & 3;  // column group: 6 tiles each
  int row0 = rowTile << 4;
  if (row0 >= M) return;

  int m = lane & 15;
  int kbh = lane >> 4;
  constexpr int KF = K / 4;

  const v2f* arow = (const v2f*)(A + (size_t)(row0 + m) * K);
  v2f af[KF];
#pragma unroll
  for (int kk = 0; kk < KF; ++kk) af[kk] = arow[2 * kk + kbh];
  int rb = kbh << 3;

  for (int tt = 0; tt < 6; ++tt) {
    int col = ((cg * 6 + tt) << 4) + m;
    const v2f* wrow = (const v2f*)(Wt + (size_t)col * K);
    v8f acc = {};
#pragma unroll
    for (int kk = 0; kk < KF; ++kk)
      acc = wmma_f32(af[kk], wrow[2 * kk + kbh], acc);
    float bv = bias[col];
#pragma unroll
    for (int v = 0; v < 8; ++v)
      Out[(size_t)(row0 + rb + v) * GATE3 + col] = acc[v] + bv;
  }
}

// ---------------------------------------------------------------------------
// One GRU time step, fused:  gh = h @ w_hh^T + b_hh  then gate math.
// Block = 256 threads (8 waves) handles 16 batch rows; 24 col tiles -> 3/wave.
// ---------------------------------------------------------------------------
#define SH_STRIDE 132  // pad 128 -> 132 floats: avoids 16-way LDS conflicts
__global__ __launch_bounds__(256) void gru_step(
    const float* __restrict__ xw,   // [BATCH x 384] for this t
    const float* __restrict__ Whh,  // [384 x HID] row-major (Bt form)
    const float* __restrict__ bhh,  // [384]
    float* __restrict__ h,          // [BATCH x HID], updated in place
    float* __restrict__ hs_out) {   // [BATCH x HID] layer output at t
  __shared__ float sh[16 * SH_STRIDE];
  __shared__ float sg[16 * GATE3];
  int row0 = blockIdx.x << 4;
  int tid = threadIdx.x;

  for (int i = tid; i < 16 * HID; i += 256) {
    int r = i >> 7, c = i & 127;
    sh[r * SH_STRIDE + c] = h[(size_t)(row0 + r) * HID + c];
  }
  __syncthreads();

  int wave = tid >> 5, lane = tid & 31;
  int m = lane & 15;
  int kbh = lane >> 4;
  const v2f* shp = (const v2f*)(sh + m * SH_STRIDE);
  v2f af[HID / 4];
#pragma unroll
  for (int kk = 0; kk < HID / 4; ++kk) af[kk] = shp[2 * kk + kbh];
  int rb = kbh << 3;

#pragma unroll
  for (int tci = 0; tci < 3; ++tci) {
    int col = ((wave * 3 + tci) << 4) + m;
    const v2f* wrow = (const v2f*)(Whh + (size_t)col * HID);
    v8f acc = {};
#pragma unroll
    for (int kk = 0; kk < HID / 4; ++kk)
      acc = wmma_f32(af[kk], wrow[2 * kk + kbh], acc);
    float bv = bhh[col];
#pragma unroll
    for (int v = 0; v < 8; ++v) sg[(rb + v) * GATE3 + col] = acc[v] + bv;
  }
  __syncthreads();

  for (int i = tid; i < 16 * HID; i += 256) {
    int r = i >> 7, j = i & 127;
    const float* xwr = xw + (size_t)(row0 + r) * GATE3;
    float rg = sigm(xwr[j] + sg[r * GATE3 + j]);
    float zg = sigm(xwr[HID + j] + sg[r * GATE3 + HID + j]);
    float ng = tanhf(xwr[2 * HID + j] + rg * sg[r * GATE3 + 2 * HID + j]);
    float hv = sh[r * SH_STRIDE + j];
    float hn = (1.0f - zg) * ng + zg * hv;
    h[(size_t)(row0 + r) * HID + j] = hn;
    hs_out[(size_t)(row0 + r) * HID + j] = hn;
  }
}

// ---------------------------------------------------------------------------
// Pooling helpers
// ---------------------------------------------------------------------------
__global__ __launch_bounds__(256) void pool_sum_vec4(
    const float* __restrict__ x, const int* __restrict__ labels,
    float* __restrict__ sums, int N, int F) {
  unsigned idx = blockIdx.x * blockDim.x + threadIdx.x;
  unsigned chunks = (unsigned)F >> 2;
  if (idx >= (unsigned)N * chunks) return;
  unsigned n = idx / chunks;
  unsigned c = idx - n * chunks;
  float4 v = *((const float4*)(x + (size_t)n * F) + c);
  float* sp = sums + (size_t)labels[n] * F + c * 4;
  atomicAdd(sp + 0, v.x);
  atomicAdd(sp + 1, v.y);
  atomicAdd(sp + 2, v.z);
  atomicAdd(sp + 3, v.w);
}

__global__ __launch_bounds__(256) void pool_cnt(const int* __restrict__ labels,
                                                float* __restrict__ cnts,
                                                int N) {
  int n = blockIdx.x * blockDim.x + threadIdx.x;
  if (n < N) atomicAdd(&cnts[labels[n]], 1.0f);
}

// seq build: xs[t][b][f] = sums[label_map[b][t]][f] / max(cnt, 1)
__global__ __launch_bounds__(256) void build_seq(
    const float* __restrict__ sums, const float* __restrict__ cnts,
    const int* __restrict__ label_map, float* __restrict__ xs, int F) {
  int idx = blockIdx.x * blockDim.x + threadIdx.x;
  int total = TSTEPS * BATCH * F;
  if (idx >= total) return;
  int f = idx % F;
  int bt = idx / F;
  int b = bt / TSTEPS;
  int t = bt - b * TSTEPS;
  int g = label_map[b * TSTEPS + t];
  float c = fmaxf(cnts[g], 1.0f);
  xs[((size_t)t * BATCH + b) * F + f] = sums[(size_t)g * F + f] / c;
}

// ---------------------------------------------------------------------------
// Decoder: out[b] = sigmoid(h[b,:] . w + b0), one wave per batch row
// ---------------------------------------------------------------------------
__global__ __launch_bounds__(256) void decoder(const float* __restrict__ h,
                                               const float* __restrict__ w,
                                               const float* __restrict__ b,
                                               float* __restrict__ out) {
  int wave = (int)((blockIdx.x * blockDim.x + threadIdx.x) >> 5);
  int lane = threadIdx.x & 31;
  if (wave >= BATCH) return;
  const float* hr = h + (size_t)wave * HID;
  float s = 0.0f;
#pragma unroll
  for (int f = lane; f < HID; f += 32) s += hr[f] * w[f];
#pragma unroll
  for (int off = 16; off > 0; off >>= 1) s += __shfl_down(s, off, 32);
  if (lane == 0) out[wave] = sigm(s + b[0]);
}

// ---------------------------------------------------------------------------
// Host orchestration
// ---------------------------------------------------------------------------
extern "C" void kernel_launch(void* const* d_in, const int* in_sizes, int n_in,
                              void* d_out, int out_size, void* d_ws,
                              size_t ws_size, hipStream_t stream) {
  (void)in_sizes; (void)n_in; (void)out_size; (void)ws_size;
  const float* x0 = (const float*)d_in[0];
  const int* ei = (const int*)d_in[1];
  const int* src = ei;
  const int* dst = ei + N_EDGES;
  const float* ew = (const float*)d_in[2];
  const int* labels = (const int*)d_in[3];
  const int* lmap = (const int*)d_in[4];
  const float* wr1 = (const float*)d_in[5];
  const float* we1 = (const float*)d_in[6];
  const float* b1 = (const float*)d_in[7];
  const float* wr2 = (const float*)d_in[8];
  const float* we2 = (const float*)d_in[9];
  const float* b2 = (const float*)d_in[10];
  const float* wr3 = (const float*)d_in[11];
  const float* we3 = (const float*)d_in[12];
  const float* b3 = (const float*)d_in[13];
  const float* wih1 = (const float*)d_in[14];
  const float* whh1 = (const float*)d_in[15];
  const float* bih1 = (const float*)d_in[16];
  const float* bhh1 = (const float*)d_in[17];
  const float* wih2 = (const float*)d_in[18];
  const float* whh2 = (const float*)d_in[19];
  const float* bih2 = (const float*)d_in[20];
  const float* bhh2 = (const float*)d_in[21];
  const float* dec_w = (const float*)d_in[22];
  const float* dec_b = (const float*)d_in[23];
  float* out = (float*)d_out;

  char* ws = (char*)d_ws;
  const size_t BUF = (size_t)N_NODES * 64 * sizeof(float);  // 64MB
  float* bufA = (float*)(ws);
  float* bufB = (float*)(ws + BUF);
  float* bufC = (float*)(ws + 2 * BUF);

  const int TB = 256;

  // ---- GCN layer 1 (pad K=5 -> 8): x0 -> bufB[N,32] ----
  float* xpad = bufA;                               // [N,8]  8 MB
  float* agg1 = (float*)((char*)bufA + (8u << 20)); // [N,8]  8 MB
  float* wr1p = (float*)((char*)bufA + (16u << 20));
  float* we1p = wr1p + 8 * 32;
  pad_x0<<<(N_NODES * 8 + TB - 1) / TB, TB, 0, stream>>>(x0, xpad, N_NODES);
  pad_w5<<<1, TB, 0, stream>>>(wr1, wr1p);
  pad_w5<<<1, TB, 0, stream>>>(we1, we1p);
  hipMemsetAsync(agg1, 0, (size_t)N_NODES * 8 * sizeof(float), stream);
  scatter_f5<<<(N_EDGES + TB - 1) / TB, TB, 0, stream>>>(xpad, src, dst, ew,
                                                         agg1, N_EDGES);
  gcn_gemm_relu_t<8, 32><<<(N_NODES / 16) / 8, TB, 0, stream>>>(
      xpad, agg1, wr1p, we1p, b1, bufB, N_NODES);

  // ---- GCN layer 2: bufB[N,32] -> bufA[N,64], agg in bufC ----
  hipMemsetAsync(bufC, 0, (size_t)N_NODES * 32 * sizeof(float), stream);
  scatter_vec4<<<((size_t)N_EDGES * 8 + TB - 1) / TB, TB, 0, stream>>>(
      bufB, src, dst, ew, bufC, N_EDGES, 32);
  gcn_gemm_relu_t<32, 64><<<(N_NODES / 16) / 8, TB, 0, stream>>>(
      bufB, bufC, wr2, we2, b2, bufA, N_NODES);

  // ---- GCN layer 3: bufA[N,64] -> bufB[N,64], agg in bufC ----
  hipMemsetAsync(bufC, 0, (size_t)N_NODES * 64 * sizeof(float), stream);
  scatter_vec4<<<((size_t)N_EDGES * 16 + TB - 1) / TB, TB, 0, stream>>>(
      bufA, src, dst, ew, bufC, N_EDGES, 64);
  gcn_gemm_relu_t<64, 64><<<(N_NODES / 16) / 8, TB, 0, stream>>>(
      bufA, bufC, wr3, we3, b3, bufB, N_NODES);

  // ---- mean pool: bufB[N,64] -> sums(bufA), cnts ----
  float* sums = bufA;
  float* cnts = (float*)((char*)bufA + (size_t)N_GRAPHS * 64 * sizeof(float));
  hipMemsetAsync(sums, 0,
                 (size_t)N_GRAPHS * 64 * sizeof(float) +
                     (size_t)N_GRAPHS * sizeof(float),
                 stream);
  pool_sum_vec4<<<((size_t)N_NODES * 16 + TB - 1) / TB, TB, 0, stream>>>(
      bufB, labels, sums, N_NODES, 64);
  pool_cnt<<<(N_NODES + TB - 1) / TB, TB, 0, stream>>>(labels, cnts, N_NODES);

  // ---- sequence build: xs0 at bufC [T,B,64] ----
  float* xs0 = bufC;
  build_seq<<<(TSTEPS * BATCH * 64 + TB - 1) / TB, TB, 0, stream>>>(
      sums, cnts, lmap, xs0, 64);

  // GRU scratch (node buffers free now except xs0 at start of bufC)
  float* xw = (float*)((char*)bufC + (4u << 20));   // 12 MB [T*B,384]
  float* hs1 = bufB;                                // 4 MB  [T,B,128]
  float* hs2 = (float*)((char*)bufB + (4u << 20));  // 4 MB
  float* h1 = (float*)((char*)bufC + (20u << 20));
  float* h2 = (float*)((char*)bufC + (21u << 20));

  const int M = TSTEPS * BATCH;  // 8192
  // ---- GRU layer 1 ----
  gru_xw_gemm<64><<<((M / 16) * 4) / 8, TB, 0, stream>>>(xs0, wih1, bih1, xw,
                                                         M);
  hipMemsetAsync(h1, 0, (size_t)BATCH * HID * sizeof(float), stream);
  for (int t = 0; t < TSTEPS; ++t) {
    gru_step<<<BATCH / 16, TB, 0, stream>>>(xw + (size_t)t * BATCH * GATE3,
                                            whh1, bhh1, h1,
                                            hs1 + (size_t)t * BATCH * HID);
  }
  // ---- GRU layer 2 ----
  gru_xw_gemm<128><<<((M / 16) * 4) / 8, TB, 0, stream>>>(hs1, wih2, bih2, xw,
                                                          M);
  hipMemsetAsync(h2, 0, (size_t)BATCH * HID * sizeof(float), stream);
  for (int t = 0; t < TSTEPS; ++t) {
    gru_step<<<BATCH / 16, TB, 0, stream>>>(xw + (size_t)t * BATCH * GATE3,
                                            whh2, bhh2, h2,
                                            hs2 + (size_t)t * BATCH * HID);
  }

  // ---- decoder: h2 -> out[512] ----
  decoder<<<(BATCH * 32 + TB - 1) / TB, TB, 0, stream>>>(h2, dec_w, dec_b,
                                                         out);
}